// BrainBlock_16904991277609
// MI455X (gfx1250) — compile-verified
//
#include <hip/hip_runtime.h>
#include <math.h>

typedef float v2f __attribute__((ext_vector_type(2)));
typedef float v8f __attribute__((ext_vector_type(8)));

#define DFEAT 128
#define NEG_SLOPE 0.01f
#define LN_EPS 1e-5f

// ---------------------------------------------------------------------------
// Kernel 1: zero the output accumulator, init degree with self-loop weight 1.0
// (must run every launch: graph replays do not re-zero workspace/output)
// ---------------------------------------------------------------------------
__global__ void init_kernel(float* __restrict__ out, float* __restrict__ deg, int N) {
    int i = blockIdx.x * blockDim.x + threadIdx.x;
    if (i < N * DFEAT) out[i] = 0.0f;
    if (i < N) deg[i] = 1.0f;  // self-loop contributes weight 1.0 to degree
}

// ---------------------------------------------------------------------------
// Kernel 2: deg[dst] += |edge_attr[e,0]|
// ---------------------------------------------------------------------------
__global__ void edge_deg_kernel(const float* __restrict__ ea, const int* __restrict__ ei,
                                float* __restrict__ deg, int E) {
    int e = blockIdx.x * blockDim.x + threadIdx.x;
    if (e < E) {
        int dst = ei[E + e];  // edge_index[1][e]
        atomicAdd(&deg[dst], fabsf(ea[(size_t)e * 4]));
    }
}

// ---------------------------------------------------------------------------
// Kernel 3: dinv = deg > 0 ? rsqrt(deg) : 0
// ---------------------------------------------------------------------------
__global__ void dinv_kernel(const float* __restrict__ deg, float* __restrict__ dinv, int N) {
    int i = blockIdx.x * blockDim.x + threadIdx.x;
    if (i < N) {
        float d = deg[i];
        dinv[i] = (d > 0.0f) ? rsqrtf(d) : 0.0f;
    }
}

// ---------------------------------------------------------------------------
// Kernel 4: h = x @ W^T + b using V_WMMA_F32_16X16X4_F32 (fp32-exact path).
// One wave32 per 16x16 output tile; K-loop of 128/4 = 32 WMMAs.
//
// ISA register layouts (CDNA5 ISA):
//   A 16x4  f32: lanes 0-15 row M=lane  {VGPR0=K0, VGPR1=K1};
//                lanes 16-31 row M=lane-16 {VGPR0=K2, VGPR1=K3}
//   B 4x16  f32: lanes 0-15 col N=lane  {K0,K1}; lanes 16-31 col N {K2,K3}
//   C/D 16x16:   VGPR v: lanes 0-15 -> M=v, N=lane; lanes 16-31 -> M=v+8
// ---------------------------------------------------------------------------
__global__ __launch_bounds__(32) void gemm_wmma_kernel(const float* __restrict__ x,
                                                       const float* __restrict__ W,
                                                       const float* __restrict__ bias,
                                                       float* __restrict__ h, int N) {
    int tile = blockIdx.x;
    int tn = tile & 7;          // 128/16 = 8 column tiles
    int tm = tile >> 3;
    int lane = threadIdx.x;
    int half = lane >> 4;       // 0: K pair {0,1}; 1: K pair {2,3}
    int l16  = lane & 15;

    int m = tm * 16 + l16;
    if (m >= N) m = N - 1;      // clamp loads for partial tail tile
    int n = tn * 16 + l16;

    const float* xrow = x + (size_t)m * DFEAT + half * 2;   // A: row m, K pairs
    const float* wrow = W + (size_t)n * DFEAT + half * 2;   // B[k][n] = W[n][k]

    v8f c = {};
    #pragma unroll 4
    for (int k0 = 0; k0 < DFEAT; k0 += 4) {
        v2f a, b;
        a.x = xrow[k0];     a.y = xrow[k0 + 1];
        b.x = wrow[k0];     b.y = wrow[k0 + 1];
        // 8 args: (neg_a, A, neg_b, B, c_mod, C, reuse_a, reuse_b)
        c = __builtin_amdgcn_wmma_f32_16x16x4_f32(false, a, false, b,
                                                  (short)0, c, false, false);
    }

    float bv = bias[tn * 16 + l16];
    int row0 = tm * 16 + half * 8;
    float* hp = h + (size_t)row0 * DFEAT + tn * 16 + l16;
    if (row0 + 8 <= N) {
        // full tile: branch-free store epilogue (always taken for N % 16 == 0)
        #pragma unroll
        for (int v = 0; v < 8; v++) {
            hp[(size_t)v * DFEAT] = c[v] + bv;
        }
    } else {
        #pragma unroll
        for (int v = 0; v < 8; v++) {
            if (row0 + v < N) hp[(size_t)v * DFEAT] = c[v] + bv;
        }
    }
}

// ---------------------------------------------------------------------------
// Kernel 5: edge scatter. One wave32 per edge; each lane owns 4 contiguous
// features -> single b128 gather per lane (wave covers the full 512B row),
// then 4 f32 atomics. out[dst] += h[src] * norm
// ---------------------------------------------------------------------------
__global__ void scatter_kernel(const float* __restrict__ h, const float* __restrict__ ea,
                               const int* __restrict__ ei, const float* __restrict__ dinv,
                               float* __restrict__ out, int E) {
    int wave = (int)((blockIdx.x * (size_t)blockDim.x + threadIdx.x) >> 5);
    int lane = threadIdx.x & 31;
    if (wave >= E) return;
    int src = ei[wave];
    int dst = ei[E + wave];
    float norm = dinv[src] * fabsf(ea[(size_t)wave * 4]) * dinv[dst];
    int d0 = lane * 4;
    const float4 hv = *(const float4*)(h + (size_t)src * DFEAT + d0);
    float* od = out + (size_t)dst * DFEAT + d0;
    atomicAdd(&od[0], hv.x * norm);
    atomicAdd(&od[1], hv.y * norm);
    atomicAdd(&od[2], hv.z * norm);
    atomicAdd(&od[3], hv.w * norm);
}

// ---------------------------------------------------------------------------
// Kernel 6: self-loop add + LeakyReLU + LayerNorm, in place on d_out.
// One wave32 per node, 4 contiguous features per lane (b128 loads/stores),
// shfl_xor reductions (wave32).
// ---------------------------------------------------------------------------
__global__ void finish_kernel(const float* __restrict__ h, const float* __restrict__ dinv,
                              const float* __restrict__ gamma, const float* __restrict__ beta,
                              float* __restrict__ out, int N) {
    int node = (int)((blockIdx.x * (size_t)blockDim.x + threadIdx.x) >> 5);
    int lane = threadIdx.x & 31;
    if (node >= N) return;

    float di = dinv[node];
    float self_norm = di * di;  // dinv[i] * 1.0 * dinv[i]

    int d0 = lane * 4;
    float4 ov = *(const float4*)(out + (size_t)node * DFEAT + d0);
    float4 hv = *(const float4*)(h + (size_t)node * DFEAT + d0);

    float v[4];
    v[0] = ov.x + hv.x * self_norm;
    v[1] = ov.y + hv.y * self_norm;
    v[2] = ov.z + hv.z * self_norm;
    v[3] = ov.w + hv.w * self_norm;

    float sum = 0.0f;
    #pragma unroll
    for (int j = 0; j < 4; j++) {
        v[j] = (v[j] >= 0.0f) ? v[j] : NEG_SLOPE * v[j];  // LeakyReLU
        sum += v[j];
    }
    #pragma unroll
    for (int off = 16; off > 0; off >>= 1) sum += __shfl_xor(sum, off, 32);
    float mean = sum * (1.0f / DFEAT);

    float vs = 0.0f;
    #pragma unroll
    for (int j = 0; j < 4; j++) {
        float dlt = v[j] - mean;
        vs += dlt * dlt;
    }
    #pragma unroll
    for (int off = 16; off > 0; off >>= 1) vs += __shfl_xor(vs, off, 32);
    float inv_std = rsqrtf(vs * (1.0f / DFEAT) + LN_EPS);

    float4 gv = *(const float4*)(gamma + d0);
    float4 bv = *(const float4*)(beta + d0);
    float4 res;
    res.x = (v[0] - mean) * inv_std * gv.x + bv.x;
    res.y = (v[1] - mean) * inv_std * gv.y + bv.y;
    res.z = (v[2] - mean) * inv_std * gv.z + bv.z;
    res.w = (v[3] - mean) * inv_std * gv.w + bv.w;
    *(float4*)(out + (size_t)node * DFEAT + d0) = res;
}

// ---------------------------------------------------------------------------
// Launch: inputs are (x, edge_attr, W, b, gamma, beta, edge_index)
// ---------------------------------------------------------------------------
extern "C" void kernel_launch(void* const* d_in, const int* in_sizes, int n_in,
                              void* d_out, int out_size, void* d_ws, size_t ws_size,
                              hipStream_t stream) {
    const float* x     = (const float*)d_in[0];
    const float* ea    = (const float*)d_in[1];
    const float* W     = (const float*)d_in[2];
    const float* bias  = (const float*)d_in[3];
    const float* gamma = (const float*)d_in[4];
    const float* beta  = (const float*)d_in[5];
    const int*   ei    = (const int*)d_in[6];   // [2, E] flat: sources then targets
    float* out = (float*)d_out;

    const int N = in_sizes[0] / DFEAT;   // 10000
    const int E = in_sizes[1] / 4;       // 640000

    // workspace layout: deg[N] | dinv[N] | h[N*128]
    float* deg  = (float*)d_ws;
    float* dinv = deg + N;
    float* h    = dinv + N;

    init_kernel<<<(N * DFEAT + 255) / 256, 256, 0, stream>>>(out, deg, N);
    edge_deg_kernel<<<(E + 255) / 256, 256, 0, stream>>>(ea, ei, deg, E);
    dinv_kernel<<<(N + 255) / 256, 256, 0, stream>>>(deg, dinv, N);

    int tiles = ((N + 15) / 16) * (DFEAT / 16);  // 625 * 8
    gemm_wmma_kernel<<<tiles, 32, 0, stream>>>(x, W, bias, h, N);

    size_t scatter_threads = (size_t)E * 32;
    scatter_kernel<<<(int)((scatter_threads + 255) / 256), 256, 0, stream>>>(h, ea, ei, dinv, out, E);

    size_t finish_threads = (size_t)N * 32;
    finish_kernel<<<(int)((finish_threads + 255) / 256), 256, 0, stream>>>(h, dinv, gamma, beta, out, N);
}